// Model_28681791602761
// MI455X (gfx1250) — compile-verified
//
#include <hip/hip_runtime.h>

// Paged-KV page-index gather (SGLang-style req_to_token -> kv_indices).
//
// Shapes fixed by the reference's setup_inputs():
//   req_to_token     : [512, 32768] int32
//   req_pool_indices : [256]        int64
//   page_kernel_lens : [256]        int64
//   kv_start_idx     : [256]        int32
//   out (kv_indices) : [256, 512]   int64
//
// Memory-bound gather: ~1.5 MB total traffic -> latency bound on MI455X
// (23.3 TB/s). No matrix math, so no WMMA path applies. Optimization is
// wave32 blocking + scalar (s_load) per-row metadata + b128 coalesced stores.

#define PAGE_SIZE 64
#define MAX_CTX   32768
#define MAX_PAGES (MAX_CTX / PAGE_SIZE)   // 512 pages per request

typedef __attribute__((ext_vector_type(2))) long long v2i64;  // 16-byte store

__global__ void __launch_bounds__(MAX_PAGES / 2)
page_index_gather(const int* __restrict__ req_to_token,
                  const long long* __restrict__ req_pool_indices,
                  const long long* __restrict__ page_kernel_lens,
                  const int* __restrict__ kv_start_idx,
                  long long* __restrict__ out)
{
    const int b = blockIdx.x;  // one block per request row

    // Wave-uniform per-row metadata -> compiler emits scalar loads.
    const long long slot = req_pool_indices[b];
    const int len        = (int)page_kernel_lens[b];
    const int kvs        = kv_start_idx[b];
    const int num_paged  = (len + PAGE_SIZE - 1) / PAGE_SIZE;

    const int* __restrict__ row = req_to_token + slot * (long long)MAX_CTX;
    long long* __restrict__ orow = out + (long long)b * MAX_PAGES;

    // Each thread handles two consecutive pages -> one b128 store.
    const int p0 = (int)threadIdx.x * 2;
    const int p1 = p0 + 1;

    int off0 = kvs + p0 * PAGE_SIZE;
    int off1 = kvs + p1 * PAGE_SIZE;
    off0 = min(max(off0, 0), MAX_CTX - 1);
    off1 = min(max(off1, 0), MAX_CTX - 1);

    // Two independent 4B gathers (256B stride) kept in flight together.
    const int d0 = row[off0];
    const int d1 = row[off1];

    // data // 64 ; arithmetic >> matches Python floor-division semantics here.
    v2i64 v;
    v.x = (p0 < num_paged) ? (long long)(d0 >> 6) : 0ll;
    v.y = (p1 < num_paged) ? (long long)(d1 >> 6) : 0ll;

    *(v2i64*)(orow + p0) = v;   // coalesced global_store_b128
}

extern "C" void kernel_launch(void* const* d_in, const int* in_sizes, int n_in,
                              void* d_out, int out_size, void* d_ws, size_t ws_size,
                              hipStream_t stream)
{
    (void)n_in; (void)out_size; (void)d_ws; (void)ws_size;

    const int*       req_to_token     = (const int*)d_in[0];
    const long long* req_pool_indices = (const long long*)d_in[1];
    const long long* page_kernel_lens = (const long long*)d_in[2];
    const int*       kv_start_idx     = (const int*)d_in[3];
    long long*       out              = (long long*)d_out;

    const int B = in_sizes[1];  // batch = 256 (size of req_pool_indices)

    dim3 grid(B);
    dim3 block(MAX_PAGES / 2);  // 256 threads = 8 wave32 per block
    page_index_gather<<<grid, block, 0, stream>>>(
        req_to_token, req_pool_indices, page_kernel_lens, kv_start_idx, out);
}